// GNNEnc_69638599737458
// MI455X (gfx1250) — compile-verified
//
#include <hip/hip_runtime.h>

typedef __attribute__((ext_vector_type(2))) float v2f;
typedef __attribute__((ext_vector_type(8))) float v8f;

// ---------------------------------------------------------------------------
// Degree / normalization
// ---------------------------------------------------------------------------
__global__ void deg_init_kernel(float* __restrict__ deg, int n) {
    int i = blockIdx.x * blockDim.x + threadIdx.x;
    if (i < n) deg[i] = 1.0f;   // self-loop contributes 1 to every node
}

__global__ void deg_accum_kernel(const int* __restrict__ dst, float* __restrict__ deg, int E) {
    int e = blockIdx.x * blockDim.x + threadIdx.x;
    if (e < E) {
        __builtin_prefetch(dst + e + 4096, 0, 0);   // global_prefetch_b8
        unsafeAtomicAdd(&deg[dst[e]], 1.0f);        // global_atomic_add_f32
    }
}

__global__ void deg_rsqrt_kernel(float* __restrict__ deg, int n) {
    int i = blockIdx.x * blockDim.x + threadIdx.x;
    if (i < n) deg[i] = rsqrtf(deg[i]);             // deg >= 1 always (self-loops)
}

__global__ void zero4_kernel(float4* __restrict__ p, int n4) {
    int i = blockIdx.x * blockDim.x + threadIdx.x;
    if (i < n4) p[i] = make_float4(0.f, 0.f, 0.f, 0.f);   // global_store_b128
}

// ---------------------------------------------------------------------------
// WMMA GEMM: C[M,NC] = A[M,K] @ B[K,NC] using V_WMMA_F32_16X16X4_F32.
// One wave per (16-row tile, 16-col tile); block has NC/16 waves.
// M must be a multiple of 16 (100000 = 6250*16). EXEC stays all-1s.
// A-frag layout (16x4 f32): lanes 0-15 hold K=0,1; lanes 16-31 hold K=2,3.
// C/D layout: VGPR r -> row r (lanes 0-15) / row r+8 (lanes 16-31), col = lane%16.
// ---------------------------------------------------------------------------
template <int K, int NC, bool BIAS_RELU>
__global__ void gemm_wmma_kernel(const float* __restrict__ A, const float* __restrict__ B,
                                 const float* __restrict__ bias, float* __restrict__ C) {
    const int lane = threadIdx.x & 31;
    const int wave = threadIdx.x >> 5;        // column tile
    const int rt   = blockIdx.x;              // row tile
    const int half = lane >> 4;               // 0 or 1
    const int l16  = lane & 15;
    const int m    = rt * 16 + l16;
    const int n    = wave * 16 + l16;

    v8f acc = {};
#pragma unroll
    for (int kb = 0; kb < K; kb += 4) {
        const int kp = kb + half * 2;
        v2f a;
        a.x = A[m * K + kp];
        a.y = A[m * K + kp + 1];
        v2f b;
        b.x = B[kp * NC + n];
        b.y = B[(kp + 1) * NC + n];
        acc = __builtin_amdgcn_wmma_f32_16x16x4_f32(
            /*neg_a=*/false, a, /*neg_b=*/false, b,
            /*c_mod=*/(short)0, acc, /*reuse_a=*/false, /*reuse_b=*/false);
    }

    const float bv = BIAS_RELU ? bias[n] : 0.0f;
#pragma unroll
    for (int r = 0; r < 8; ++r) {
        const int row = rt * 16 + r + half * 8;
        float v = acc[r];
        if (BIAS_RELU) {
            v += bv;
            v = v > 0.0f ? v : 0.0f;
        }
        C[row * NC + n] = v;
    }
}

// ---------------------------------------------------------------------------
// Edge propagation: agg[dst] += h[src] * dis[src]*dis[dst]
// Thread = (edge, 4-feature group): one b128 gather of H + 4 hw fp32 atomics.
// G = F/4 consecutive threads share one edge -> index loads hit the same
// cacheline; feature accesses are coalesced. Atomics land in L2 (tables fit
// easily in the 192 MB L2).
// ---------------------------------------------------------------------------
template <int F>
__global__ void propagate_kernel(const int* __restrict__ src, const int* __restrict__ dst,
                                 const float* __restrict__ dis, const float* __restrict__ H,
                                 float* __restrict__ agg, int E) {
    constexpr int G = F / 4;                  // 4-float groups per row
    unsigned gid = blockIdx.x * blockDim.x + threadIdx.x;
    int e = (int)(gid / G);
    int f = (int)(gid % G) * 4;
    if (e >= E) return;
    __builtin_prefetch(src + e + 2048, 0, 0); // global_prefetch_b8
    __builtin_prefetch(dst + e + 2048, 0, 0);
    int s = src[e];
    int d = dst[e];
    float nrm = dis[s] * dis[d];
    const float4 hv = *(const float4*)(H + (size_t)s * F + f);  // global_load_b128
    float* ap = agg + (size_t)d * F + f;
    unsafeAtomicAdd(ap + 0, hv.x * nrm);      // global_atomic_add_f32 x4
    unsafeAtomicAdd(ap + 1, hv.y * nrm);
    unsafeAtomicAdd(ap + 2, hv.z * nrm);
    unsafeAtomicAdd(ap + 3, hv.w * nrm);
}

// ---------------------------------------------------------------------------
// out = relu(agg + h*dis^2 + bias)   (self-loop term folded in; in-place safe)
// ---------------------------------------------------------------------------
template <int F>
__global__ void selfloop_bias_relu_kernel(const float* __restrict__ agg, const float* __restrict__ H,
                                          const float* __restrict__ dis, const float* __restrict__ bias,
                                          float* __restrict__ out, int total) {
    int i = blockIdx.x * blockDim.x + threadIdx.x;
    if (i >= total) return;
    int node = i / F;
    int f    = i % F;
    float ds = dis[node];
    float v  = agg[i] + H[i] * ds * ds + bias[f];
    out[i]   = v > 0.0f ? v : 0.0f;
}

// ---------------------------------------------------------------------------
// Launch
// ---------------------------------------------------------------------------
extern "C" void kernel_launch(void* const* d_in, const int* in_sizes, int n_in,
                              void* d_out, int out_size, void* d_ws, size_t ws_size,
                              hipStream_t stream) {
    const float* x   = (const float*)d_in[0];
    const int*   ei  = (const int*)d_in[1];   // [2, E] int32
    const float* W1  = (const float*)d_in[2]; // [4, 64]
    const float* b1  = (const float*)d_in[3]; // [64]
    const float* W2  = (const float*)d_in[4]; // [64, 32]
    const float* b2  = (const float*)d_in[5]; // [32]
    const float* Wfc = (const float*)d_in[6]; // [32, 32]
    const float* bfc = (const float*)d_in[7]; // [32]
    float* out = (float*)d_out;

    const int N = in_sizes[0] / 4;   // 100000
    const int E = in_sizes[1] / 2;   // 3200000
    const int* srcIdx = ei;
    const int* dstIdx = ei + E;

    // Workspace layout (floats): dis[N] | h1[N*64] | buf2[N*64]
    float* ws   = (float*)d_ws;
    float* dis  = ws;
    float* h1   = ws + (((size_t)N + 255) & ~(size_t)255);  // 1 KB-aligned region
    float* buf2 = h1 + (size_t)N * 64;
    float* agg1 = buf2;                   // N*64 (dead after epilogue 1)
    float* h2   = buf2;                   // N*32 (reuses agg1 space)
    float* agg2 = buf2 + (size_t)N * 32;  // N*32
    float* h2o  = h1;                     // N*32 (h1 dead after layer-2 GEMM input use)

    const int T = 256;
    const int rowTiles = N / 16;          // 6250 (exact)

    // 1) degree -> dis
    deg_init_kernel<<<(N + T - 1) / T, T, 0, stream>>>(dis, N);
    deg_accum_kernel<<<(E + T - 1) / T, T, 0, stream>>>(dstIdx, dis, E);
    deg_rsqrt_kernel<<<(N + T - 1) / T, T, 0, stream>>>(dis, N);

    // 2) h1 = x @ W1   (K=4, NC=64 -> 4 waves/block, 1 WMMA each)
    gemm_wmma_kernel<4, 64, false><<<rowTiles, 128, 0, stream>>>(x, W1, nullptr, h1);

    // 3) layer-1 propagate + epilogue
    zero4_kernel<<<((N * 16) + T - 1) / T, T, 0, stream>>>((float4*)agg1, N * 16);
    {
        unsigned total = (unsigned)E * 16u;   // E * (64/4)
        propagate_kernel<64><<<(total + T - 1) / T, T, 0, stream>>>(srcIdx, dstIdx, dis, h1, agg1, E);
    }
    selfloop_bias_relu_kernel<64><<<((N * 64) + T - 1) / T, T, 0, stream>>>(agg1, h1, dis, b1, h1, N * 64);

    // 4) h2 = h1' @ W2  (K=64, NC=32 -> 2 waves/block, 16 chained WMMAs)
    gemm_wmma_kernel<64, 32, false><<<rowTiles, 64, 0, stream>>>(h1, W2, nullptr, h2);

    // 5) layer-2 propagate + epilogue
    zero4_kernel<<<((N * 8) + T - 1) / T, T, 0, stream>>>((float4*)agg2, N * 8);
    {
        unsigned total = (unsigned)E * 8u;    // E * (32/4)
        propagate_kernel<32><<<(total + T - 1) / T, T, 0, stream>>>(srcIdx, dstIdx, dis, h2, agg2, E);
    }
    selfloop_bias_relu_kernel<32><<<((N * 32) + T - 1) / T, T, 0, stream>>>(agg2, h2, dis, b2, h2o, N * 32);

    // 6) out = relu(h2' @ Wfc + bfc)  (K=32, NC=32, fused epilogue)
    gemm_wmma_kernel<32, 32, true><<<rowTiles, 64, 0, stream>>>(h2o, Wfc, bfc, out);
}